// GraphConv1TKP_40535901339794
// MI455X (gfx1250) — compile-verified
//
#include <hip/hip_runtime.h>
#include <math.h>

typedef float v2f __attribute__((ext_vector_type(2)));
typedef float v8f __attribute__((ext_vector_type(8)));

#define BGRAPH 16
#define NNODE  2048
#define NB     32768      // BGRAPH*NNODE
#define FIN    64
#define HID    128
#define KKEEP  1639       // ceil(0.8*2048)
#define BK     26224      // BGRAPH*KKEEP

__device__ __forceinline__ void atomAddF(float* p, float v) {
  __hip_atomic_fetch_add(p, v, __ATOMIC_RELAXED, __HIP_MEMORY_SCOPE_AGENT);
}

// ---------------- utility fills ----------------
__global__ void fill_f32(float* __restrict__ p, long long n, float v) {
  long long i = (long long)blockIdx.x * blockDim.x + threadIdx.x;
  if (i < n) p[i] = v;
}
__global__ void fill_i32(int* __restrict__ p, int n, int v) {
  int i = blockIdx.x * blockDim.x + threadIdx.x;
  if (i < n) p[i] = v;
}

// ---------------- edge scatter: agg[dst] += x[src] ----------------
__global__ void scatter_add(const float* __restrict__ x, const int* __restrict__ src,
                            const int* __restrict__ dst, float* __restrict__ agg,
                            int E, int F) {
  long long t = (long long)blockIdx.x * blockDim.x + threadIdx.x;
  int chunks = F >> 2;
  long long total = (long long)E * chunks;
  if (t >= total) return;
  int e = (int)(t / chunks);
  int c = (int)(t % chunks);
  const float4 v = *(const float4*)(x + (size_t)src[e] * F + c * 4);
  float* p = agg + (size_t)dst[e] * F + c * 4;
  atomAddF(p + 0, v.x); atomAddF(p + 1, v.y);
  atomAddF(p + 2, v.z); atomAddF(p + 3, v.w);
}

// masked variant for pooled graph (layer 3)
__global__ void scatter_add_masked(const float* __restrict__ hp, const int* __restrict__ src,
                                   const int* __restrict__ dst, const int* __restrict__ new_id,
                                   float* __restrict__ agg, int E) {
  long long t = (long long)blockIdx.x * blockDim.x + threadIdx.x;
  long long total = (long long)E * 32;   // F = 128
  if (t >= total) return;
  int e = (int)(t >> 5);
  int c = (int)(t & 31);
  int ns = new_id[src[e]];
  int nd = new_id[dst[e]];
  if (ns < 0 || nd < 0) return;
  const float4 v = *(const float4*)(hp + (size_t)ns * HID + c * 4);
  float* p = agg + (size_t)nd * HID + c * 4;
  atomAddF(p + 0, v.x); atomAddF(p + 1, v.y);
  atomAddF(p + 2, v.z); atomAddF(p + 3, v.w);
}

// ---------------- fused GraphConv GEMM via WMMA f32 16x16x4 ----------------
// out[m][n] = relu( sum_k A[m][k]*Wrel[n][k] + sum_k X[m][k]*Wroot[n][k] + bias[n] )
// N fixed = 128. One wave computes one 16x16 tile; 4 waves/block; grid.y*4+wid = n-tile.
__global__ void gconv_gemm_wmma(const float* __restrict__ A, int lda,
                                const float* __restrict__ X, int ldx,
                                const float* __restrict__ Wrel,
                                const float* __restrict__ Wroot,
                                const float* __restrict__ bias,
                                float* __restrict__ out,
                                int M, int Kdim) {
  const int lane = threadIdx.x & 31;
  const int wid  = threadIdx.x >> 5;
  const int m  = lane & 15;
  const int hi = lane >> 4;               // 0: lanes 0-15, 1: lanes 16-31
  const int row0 = blockIdx.x * 16;
  const int col0 = (blockIdx.y * 4 + wid) * 16;

  const int ar = min(row0 + m, M - 1);
  const float* Arow = A + (size_t)ar * lda;
  const float* Xrow = X + (size_t)ar * ldx;
  const float* Wr = Wrel  + (size_t)(col0 + m) * Kdim;  // B[k][n] = W[n][k], n = lane%16
  const float* Wt = Wroot + (size_t)(col0 + m) * Kdim;

  v8f acc = {};
  for (int k0 = 0; k0 < Kdim; k0 += 4) {
    const int ka = k0 + 2 * hi;           // v0 holds K=2*hi, v1 holds K=2*hi+1
    v2f a, xx, br, bt;
    a.x  = Arow[ka]; a.y  = Arow[ka + 1];
    xx.x = Xrow[ka]; xx.y = Xrow[ka + 1];
    br.x = Wr[ka];   br.y = Wr[ka + 1];
    bt.x = Wt[ka];   bt.y = Wt[ka + 1];
    acc = __builtin_amdgcn_wmma_f32_16x16x4_f32(false, a,  false, br, (short)0, acc, false, false);
    acc = __builtin_amdgcn_wmma_f32_16x16x4_f32(false, xx, false, bt, (short)0, acc, false, false);
  }

  const float bc = bias[col0 + m];
#pragma unroll
  for (int j = 0; j < 8; j++) {
    const int row = row0 + j + 8 * hi;    // C/D: VGPR j -> row j (+8 for high lanes)
    if (row < M) {
      float v = acc[j] + bc;
      out[(size_t)row * HID + (col0 + m)] = v > 0.f ? v : 0.f;
    }
  }
}

// ---------------- per-graph global max+mean pools -> [16,256] ----------------
__global__ void pools_kernel(const float* __restrict__ h, int Ng, float* __restrict__ out) {
  const int g = blockIdx.x;
  const int f = threadIdx.x;              // 128 threads
  const float* base = h + (size_t)g * Ng * HID + f;
  float mx = -3.402823e38f, sm = 0.f;
  for (int i = 0; i < Ng; i++) {
    float v = base[(size_t)i * HID];
    mx = fmaxf(mx, v); sm += v;
  }
  out[g * 256 + f]       = mx;
  out[g * 256 + 128 + f] = sm / (float)Ng;
}

// ---------------- pool_w inverse norm ----------------
__global__ void norm_kernel(const float* __restrict__ w, float* __restrict__ inv_norm) {
  __shared__ float red[128];
  int t = threadIdx.x;
  red[t] = w[t] * w[t];
  __syncthreads();
  for (int s = 64; s > 0; s >>= 1) {
    if (t < s) red[t] += red[t + s];
    __syncthreads();
  }
  if (t == 0) inv_norm[0] = 1.0f / sqrtf(red[0]);
}

// ---------------- score = tanh(h.pool_w / ||pool_w||) ----------------
__global__ void score_kernel(const float* __restrict__ h, const float* __restrict__ w,
                             const float* __restrict__ inv_norm, float* __restrict__ score, int n) {
  int i = blockIdx.x * blockDim.x + threadIdx.x;
  if (i >= n) return;
  const float* r = h + (size_t)i * HID;
  float s = 0.f;
  for (int k = 0; k < HID; k++) s += r[k] * w[k];
  score[i] = tanhf(s * inv_norm[0]);
}

// ---------------- per-graph bitonic top-K (descending sort of 2048) ----------------
__global__ void topk_kernel(const float* __restrict__ score, float* __restrict__ tops,
                            int* __restrict__ perm, int* __restrict__ new_id) {
  __shared__ float sv[NNODE];
  __shared__ int   si[NNODE];
  const int g = blockIdx.x, t = threadIdx.x;
  for (int i = t; i < NNODE; i += blockDim.x) { sv[i] = score[g * NNODE + i]; si[i] = i; }
  __syncthreads();
  for (int k = 2; k <= NNODE; k <<= 1) {
    for (int j = k >> 1; j > 0; j >>= 1) {
      for (int i = t; i < NNODE; i += blockDim.x) {
        int ixj = i ^ j;
        if (ixj > i) {
          bool up = ((i & k) == 0);
          float a = sv[i], b = sv[ixj];
          bool sw = up ? (a < b) : (a > b);   // descending overall
          if (sw) { sv[i] = b; sv[ixj] = a; int tmp = si[i]; si[i] = si[ixj]; si[ixj] = tmp; }
        }
      }
      __syncthreads();
    }
  }
  for (int i = t; i < KKEEP; i += blockDim.x) {
    int old = g * NNODE + si[i];
    int nid = g * KKEEP + i;
    tops[nid]   = sv[i];
    perm[nid]   = old;
    new_id[old] = nid;
  }
}

// ---------------- hp[j] = h[perm[j]] * tops[j] ----------------
__global__ void gather_scale(const float* __restrict__ h, const int* __restrict__ perm,
                             const float* __restrict__ tops, float* __restrict__ hp) {
  int t = blockIdx.x * blockDim.x + threadIdx.x;
  if (t >= BK * HID) return;
  int j = t >> 7, f = t & 127;
  hp[t] = h[(size_t)perm[j] * HID + f] * tops[j];
}

// ---------------- MLP head + log_softmax (tiny, single block) ----------------
__global__ void mlp_kernel(const float* __restrict__ x1, const float* __restrict__ x2,
                           const float* __restrict__ x3,
                           const float* __restrict__ Wl1, const float* __restrict__ bl1,
                           const float* __restrict__ Wl2, const float* __restrict__ bl2,
                           const float* __restrict__ Wl3, const float* __restrict__ bl3,
                           float* __restrict__ out) {
  __shared__ float z [16 * 256];
  __shared__ float z1[16 * 128];
  __shared__ float z2[16 * 64];
  __shared__ float z3[16 * 16];
  const int t = threadIdx.x;  // 256
  for (int i = t; i < 16 * 256; i += 256) z[i] = x1[i] + x2[i] + x3[i];
  __syncthreads();
  for (int i = t; i < 16 * 128; i += 256) {
    int m = i >> 7, n = i & 127;
    float s = bl1[n];
    const float* w = Wl1 + n * 256;
    const float* zz = z + m * 256;
    for (int k = 0; k < 256; k++) s += zz[k] * w[k];
    z1[i] = s > 0.f ? s : 0.f;
  }
  __syncthreads();
  for (int i = t; i < 16 * 64; i += 256) {
    int m = i >> 6, n = i & 63;
    float s = bl2[n];
    for (int k = 0; k < 128; k++) s += z1[m * 128 + k] * Wl2[n * 128 + k];
    z2[i] = s > 0.f ? s : 0.f;
  }
  __syncthreads();
  for (int i = t; i < 16 * 10; i += 256) {
    int m = i / 10, n = i % 10;
    float s = bl3[n];
    for (int k = 0; k < 64; k++) s += z2[m * 64 + k] * Wl3[n * 64 + k];
    z3[m * 16 + n] = s;
  }
  __syncthreads();
  if (t < 16) {
    float mx = -3.402823e38f;
    for (int c = 0; c < 10; c++) mx = fmaxf(mx, z3[t * 16 + c]);
    float se = 0.f;
    for (int c = 0; c < 10; c++) se += expf(z3[t * 16 + c] - mx);
    float lse = logf(se) + mx;
    for (int c = 0; c < 10; c++) out[t * 10 + c] = z3[t * 16 + c] - lse;
  }
}

extern "C" void kernel_launch(void* const* d_in, const int* in_sizes, int n_in,
                              void* d_out, int out_size, void* d_ws, size_t ws_size,
                              hipStream_t stream) {
  const float* x       = (const float*)d_in[0];
  const int*   src     = (const int*)  d_in[1];
  const int*   dst     = (const int*)  d_in[2];
  const float* W1_rel  = (const float*)d_in[3];
  const float* b1      = (const float*)d_in[4];
  const float* W1_root = (const float*)d_in[5];
  const float* W2_rel  = (const float*)d_in[6];
  const float* b2      = (const float*)d_in[7];
  const float* W2_root = (const float*)d_in[8];
  const float* W3_rel  = (const float*)d_in[9];
  const float* b3      = (const float*)d_in[10];
  const float* W3_root = (const float*)d_in[11];
  const float* pool_w  = (const float*)d_in[12];
  const float* Wl1     = (const float*)d_in[13];
  const float* bl1     = (const float*)d_in[14];
  const float* Wl2     = (const float*)d_in[15];
  const float* bl2     = (const float*)d_in[16];
  const float* Wl3     = (const float*)d_in[17];
  const float* bl3     = (const float*)d_in[18];
  const int E = in_sizes[1];

  float* ws = (float*)d_ws;
  size_t o = 0;
  float* agg   = ws + o; o += (size_t)NB * HID;
  float* hA    = ws + o; o += (size_t)NB * HID;   // h1, later hp (pooled feats)
  float* hB    = ws + o; o += (size_t)NB * HID;   // h2, later h3
  float* score = ws + o; o += NB;
  float* tops  = ws + o; o += BK;
  int*   perm  = (int*)(ws + o); o += BK;
  int*   nid   = (int*)(ws + o); o += NB;
  float* x1    = ws + o; o += 16 * 256;
  float* x2    = ws + o; o += 16 * 256;
  float* x3    = ws + o; o += 16 * 256;
  float* invn  = ws + o; o += 1;

  const int T = 256;
  // ---- GraphConv1 (F_IN=64 -> 128) ----
  fill_f32<<<(NB * FIN + T - 1) / T, T, 0, stream>>>(agg, (long long)NB * FIN, 0.f);
  scatter_add<<<(unsigned)(((long long)E * (FIN / 4) + T - 1) / T), T, 0, stream>>>(x, src, dst, agg, E, FIN);
  gconv_gemm_wmma<<<dim3(NB / 16, 2), 128, 0, stream>>>(agg, FIN, x, FIN, W1_rel, W1_root, b1, hA, NB, FIN);
  pools_kernel<<<BGRAPH, 128, 0, stream>>>(hA, NNODE, x1);

  // ---- GraphConv2 (128 -> 128) ----
  fill_f32<<<(NB * HID + T - 1) / T, T, 0, stream>>>(agg, (long long)NB * HID, 0.f);
  scatter_add<<<(unsigned)(((long long)E * (HID / 4) + T - 1) / T), T, 0, stream>>>(hA, src, dst, agg, E, HID);
  gconv_gemm_wmma<<<dim3(NB / 16, 2), 128, 0, stream>>>(agg, HID, hA, HID, W2_rel, W2_root, b2, hB, NB, HID);

  // ---- TopK pooling ----
  norm_kernel<<<1, 128, 0, stream>>>(pool_w, invn);
  score_kernel<<<NB / T, T, 0, stream>>>(hB, pool_w, invn, score, NB);
  fill_i32<<<NB / T, T, 0, stream>>>(nid, NB, -1);
  topk_kernel<<<BGRAPH, 256, 0, stream>>>(score, tops, perm, nid);
  gather_scale<<<(BK * HID + T - 1) / T, T, 0, stream>>>(hB, perm, tops, hA);
  pools_kernel<<<BGRAPH, 128, 0, stream>>>(hA, KKEEP, x2);

  // ---- GraphConv3 on pooled graph (masked edges) ----
  fill_f32<<<((long long)BK * HID + T - 1) / T, T, 0, stream>>>(agg, (long long)BK * HID, 0.f);
  scatter_add_masked<<<(unsigned)(((long long)E * (HID / 4) + T - 1) / T), T, 0, stream>>>(hA, src, dst, nid, agg, E);
  gconv_gemm_wmma<<<dim3(BK / 16, 2), 128, 0, stream>>>(agg, HID, hA, HID, W3_rel, W3_root, b3, hB, BK, HID);
  pools_kernel<<<BGRAPH, 128, 0, stream>>>(hB, KKEEP, x3);

  // ---- MLP head + log_softmax ----
  mlp_kernel<<<1, 256, 0, stream>>>(x1, x2, x3, Wl1, bl1, Wl2, bl2, Wl3, bl3, (float*)d_out);
}